// Step3p5MoEMLP_6674379178224
// MI455X (gfx1250) — compile-verified
//
#include <hip/hip_runtime.h>

// Problem constants (match reference)
#define T_TOK 1024
#define H_DIM 2048
#define E_NUM 16
#define I_DIM 1024
#define GU_LD (2 * I_DIM)   // gate_up_proj row stride
#define XS_LD 260           // 256 + 4 pad: rows land on distinct LDS banks
#define M_TILE 32           // tokens per block (2 WMMA m-tiles): 2x B-fragment reuse

typedef __attribute__((ext_vector_type(2))) float v2f;
typedef __attribute__((ext_vector_type(8))) float v8f;
typedef __attribute__((ext_vector_type(4))) unsigned int u32x4;
typedef __attribute__((ext_vector_type(8))) int i32x8;
typedef __attribute__((ext_vector_type(4))) int i32x4;

#if defined(__has_builtin)
#if __has_builtin(__builtin_amdgcn_tensor_load_to_lds)
#define USE_TDM 1
#endif
#endif
#ifndef USE_TDM
#define USE_TDM 0
#endif

__device__ __forceinline__ void wait_tensorcnt0() {
#if defined(__has_builtin) && __has_builtin(__builtin_amdgcn_s_wait_tensorcnt)
  __builtin_amdgcn_s_wait_tensorcnt(0);
#else
  asm volatile("s_wait_tensorcnt 0x0" ::: "memory");
#endif
}

// ---- WMMA f32 16x16x4: A 16x4 (2 VGPRs), B 4x16 (2 VGPRs), C/D 16x16 (8 VGPRs)
__device__ __forceinline__ v8f wmma4(v2f a, v2f b, v8f c) {
  return __builtin_amdgcn_wmma_f32_16x16x4_f32(
      /*neg_a=*/false, a, /*neg_b=*/false, b,
      /*c_mod=*/(short)0, c, /*reuse_a=*/false, /*reuse_b=*/false);
}

// B fragment at (k, n) of row-major W with leading dim ld.
// VGPR0 = rows k+0 (lanes 0-15) / k+2 (lanes 16-31); VGPR1 = k+1 / k+3; col = lane%16.
__device__ __forceinline__ v2f load_b_frag(const float* __restrict__ W, int ld,
                                           int k, int n) {
  const int lane = threadIdx.x & 31;
  const int c = lane & 15;
  const int r = (lane >> 4) << 1;
  const float* p = W + (size_t)(k + r) * ld + n + c;
  v2f b;
  b.x = p[0];
  b.y = p[ld];
  return b;
}

#if USE_TDM
// TDM gather-mode load: 16 rows (16-bit indices ts[r0+j]>>shift) x 256 f32 chunk
// from gaddr (tensor row stride = `stride` elems) into LDS at ldsaddr, with
// 4-dword pad every 256 dwords (matches XS_LD = 260).
__device__ __forceinline__ void tdm_gather16(const void* gaddr, unsigned int ldsaddr,
                                             unsigned int dim0, unsigned int dim1,
                                             unsigned int stride,
                                             const int* __restrict__ ts, int r0,
                                             int shift) {
  unsigned long long ga = (unsigned long long)(size_t)gaddr;
  u32x4 g0;
  g0[0] = 0x80000001u;                                   // gather_mode=1 | count=1
  g0[1] = ldsaddr;                                       // LDS byte address
  g0[2] = (unsigned int)(ga & 0xFFFFFFFFu);              // global_addr[31:0]
  g0[3] = (unsigned int)((ga >> 32) & 0x01FFFFFFu) | (2u << 30);  // [56:32] | type=2
  i32x8 g1;
  g1[0] = (2 << 16)      // data_size = 4 bytes
        | (1 << 20)      // pad_enable
        | (7 << 22)      // pad_interval: 256 dwords
        | (3 << 25);     // pad_amount: 4 dwords
  g1[1] = (int)((dim0 & 0xFFFFu) << 16);                 // tensor_dim0[15:0]
  g1[2] = (int)((dim0 >> 16) | ((dim1 & 0xFFFFu) << 16));// dim0[31:16] | dim1[15:0]
  g1[3] = (int)((dim1 >> 16) | (256u << 16));            // dim1[31:16] | tile_dim0=256
  g1[4] = 16;                                            // tile_dim1 = 16 indices
  g1[5] = (int)stride;                                   // tensor_dim0_stride[31:0]
  g1[6] = 0;
  g1[7] = 0;
  int id[16];
#pragma unroll
  for (int j = 0; j < 16; ++j) id[j] = ts[r0 + j] >> shift;
  i32x4 g2, g3;
  g2[0] = id[0] | (id[1] << 16);  g2[1] = id[2]  | (id[3] << 16);
  g2[2] = id[4] | (id[5] << 16);  g2[3] = id[6]  | (id[7] << 16);
  g3[0] = id[8] | (id[9] << 16);  g3[1] = id[10] | (id[11] << 16);
  g3[2] = id[12] | (id[13] << 16); g3[3] = id[14] | (id[15] << 16);
  i32x8 g4 = {};                                         // unused trailing group
  __builtin_amdgcn_tensor_load_to_lds(g0, g1, g2, g3, g4, 0);
}
#endif

// ---------------------------------------------------------------- init
__global__ void init_kernel(int* __restrict__ cnt) {
  if (threadIdx.x < E_NUM) cnt[threadIdx.x] = 0;
}

// ---------------------------------------------------------------- router
__global__ void router_kernel(const float* __restrict__ x,
                              const float* __restrict__ gw,
                              int* __restrict__ cnt,
                              int* __restrict__ toklist,
                              float* __restrict__ twlist) {
  const int t = blockIdx.x;
  const int tid = threadIdx.x;
  const int e = tid >> 4, j = tid & 15;
  __shared__ float red[256];
  __shared__ float ps[E_NUM];

  const float4* xr = (const float4*)(x + (size_t)t * H_DIM + j * 128);
  const float4* wr = (const float4*)(gw + (size_t)e * H_DIM + j * 128);
  float s = 0.f;
#pragma unroll 4
  for (int i = 0; i < 32; ++i) {
    float4 a = xr[i], b = wr[i];
    s += a.x * b.x + a.y * b.y + a.z * b.z + a.w * b.w;
  }
  red[tid] = s;
  __syncthreads();
  if (tid < E_NUM) {
    float l = 0.f;
    for (int i = 0; i < 16; ++i) l += red[tid * 16 + i];
    ps[tid] = 1.0f / (1.0f + __expf(-l));   // sigmoid (pre-norm cancels in renorm)
  }
  __syncthreads();
  if (tid == 0) {
    int i1 = 0; float p1 = ps[0];
    for (int i = 1; i < E_NUM; ++i) if (ps[i] > p1) { p1 = ps[i]; i1 = i; }
    int i2 = -1; float p2 = -1.f;
    for (int i = 0; i < E_NUM; ++i)
      if (i != i1 && ps[i] > p2) { p2 = ps[i]; i2 = i; }
    const float inv = 1.0f / (p1 + p2);
    int pos = atomicAdd(&cnt[i1], 1);
    toklist[i1 * T_TOK + pos] = t * 2 + 0;
    twlist[i1 * T_TOK + pos] = p1 * inv;
    pos = atomicAdd(&cnt[i2], 1);
    toklist[i2 * T_TOK + pos] = t * 2 + 1;
    twlist[i2 * T_TOK + pos] = p2 * inv;
  }
}

// ---------------------------------------------------------------- gate_up + SwiGLU
// Block = (expert, 32 gathered tokens) = 2 m-tiles; 8 waves.
// Pass p: wave w covers gate cols [p*256 + w*32, +32) and matching up cols (+I).
__global__ void moe_gu_kernel(const float* __restrict__ x,
                              const float* __restrict__ wgu,
                              const int* __restrict__ cnt,
                              const int* __restrict__ toklist,
                              float* __restrict__ inter) {
  const int e = blockIdx.y;
  const int tile0 = blockIdx.x * M_TILE;
  const int n = cnt[e];
  if (tile0 >= n) return;

  __shared__ float Xs[M_TILE * XS_LD];
  __shared__ int ts_ent[M_TILE];
  __shared__ int ts_val[M_TILE];

  const int tid = threadIdx.x;
  if (tid < M_TILE) {
    const int ok = (tile0 + tid) < n;
    ts_ent[tid] = toklist[e * T_TOK + (ok ? tile0 + tid : tile0)];
    ts_val[tid] = ok;
  }
  __syncthreads();

  const float* We = wgu + (size_t)e * H_DIM * GU_LD;
  const int w = tid >> 5, lane = tid & 31;
  const int lm = lane & 15, lk = (lane >> 4) << 1, hi = lane >> 4;
#if USE_TDM
  const unsigned int xs_lds = (unsigned int)(unsigned long long)(size_t)&Xs[0];
#endif

  for (int p = 0; p < 4; ++p) {
    v8f gacc[2][2] = {}; v8f uacc[2][2] = {};
    const int cbase = p * 256 + w * 32;
    for (int kc = 0; kc < H_DIM; kc += 256) {
      __syncthreads();                       // protect Xs reuse
#if USE_TDM
      if (tid < 32) {                        // wave 0 drives the TDM
        tdm_gather16(x + kc, xs_lds, H_DIM, T_TOK, H_DIM, ts_ent, 0, 1);
        tdm_gather16(x + kc, xs_lds + 16 * XS_LD * 4, H_DIM, T_TOK, H_DIM,
                     ts_ent, 16, 1);
        wait_tensorcnt0();
      }
#else
      {
        const int row = tid >> 3, cb = (tid & 7) * 32;
        const float4* src =
            (const float4*)(x + (size_t)(ts_ent[row] >> 1) * H_DIM + kc + cb);
        float4* dst = (float4*)(&Xs[row * XS_LD + cb]);
#pragma unroll
        for (int q = 0; q < 8; ++q) dst[q] = src[q];
        if (kc + 256 < H_DIM)
          __builtin_prefetch(x + (size_t)(ts_ent[row] >> 1) * H_DIM + kc + 256 + cb, 0, 0);
      }
#endif
      __syncthreads();
      for (int kk = 0; kk < 256; kk += 4) {
        const float* a0p = &Xs[lm * XS_LD + kk + lk];
        const float* a1p = &Xs[(lm + 16) * XS_LD + kk + lk];
        v2f a0; a0.x = a0p[0]; a0.y = a0p[1];
        v2f a1; a1.x = a1p[0]; a1.y = a1p[1];
        const int kg = kc + kk;
#pragma unroll
        for (int nt = 0; nt < 2; ++nt) {
          const int cg = cbase + nt * 16;
          v2f bg = load_b_frag(We, GU_LD, kg, cg);
          v2f bu = load_b_frag(We, GU_LD, kg, cg + I_DIM);
          gacc[0][nt] = wmma4(a0, bg, gacc[0][nt]);
          gacc[1][nt] = wmma4(a1, bg, gacc[1][nt]);
          uacc[0][nt] = wmma4(a0, bu, uacc[0][nt]);
          uacc[1][nt] = wmma4(a1, bu, uacc[1][nt]);
        }
      }
    }
#pragma unroll
    for (int mt = 0; mt < 2; ++mt) {
#pragma unroll
      for (int v = 0; v < 8; ++v) {
        const int m = mt * 16 + v + (hi << 3);
        if (ts_val[m]) {
          float* op = inter + (size_t)ts_ent[m] * I_DIM;
#pragma unroll
          for (int nt = 0; nt < 2; ++nt) {
            const float g = gacc[mt][nt][v], u = uacc[mt][nt][v];
            op[cbase + nt * 16 + lm] = (g / (1.0f + __expf(-g))) * u;
          }
        }
      }
    }
  }
}

// ---------------------------------------------------------------- down proj
// M=32 (2 m-tiles), K=I, N=H; inter rows staged to LDS (TDM gather on entries);
// unique writer per (token,slot) row -> plain stores, deterministic.
__global__ void moe_down_kernel(const float* __restrict__ inter,
                                const float* __restrict__ wd,
                                const int* __restrict__ cnt,
                                const int* __restrict__ toklist,
                                const float* __restrict__ twlist,
                                float* __restrict__ ypart) {
  const int e = blockIdx.y;
  const int tile0 = blockIdx.x * M_TILE;
  const int n = cnt[e];
  if (tile0 >= n) return;

  __shared__ float Is[M_TILE * XS_LD];
  __shared__ int ts_ent[M_TILE];
  __shared__ float ts_w[M_TILE];
  __shared__ int ts_val[M_TILE];

  const int tid = threadIdx.x;
  if (tid < M_TILE) {
    const int ok = (tile0 + tid) < n;
    const int src = e * T_TOK + (ok ? tile0 + tid : tile0);
    ts_ent[tid] = toklist[src];
    ts_w[tid] = ok ? twlist[src] : 0.f;
    ts_val[tid] = ok;
  }
  __syncthreads();

  const float* Wd = wd + (size_t)e * I_DIM * H_DIM;
  const int w = tid >> 5, lane = tid & 31;
  const int lm = lane & 15, lk = (lane >> 4) << 1, hi = lane >> 4;
#if USE_TDM
  const unsigned int is_lds = (unsigned int)(unsigned long long)(size_t)&Is[0];
#endif

  for (int np = 0; np < 4; ++np) {
    v8f acc[2][4] = {};
    const int cbase = w * 256 + np * 64;
    for (int kc = 0; kc < I_DIM; kc += 256) {
      __syncthreads();
#if USE_TDM
      if (tid < 32) {
        tdm_gather16(inter + kc, is_lds, I_DIM, 2 * T_TOK, I_DIM, ts_ent, 0, 0);
        tdm_gather16(inter + kc, is_lds + 16 * XS_LD * 4, I_DIM, 2 * T_TOK, I_DIM,
                     ts_ent, 16, 0);
        wait_tensorcnt0();
      }
#else
      {
        const int row = tid >> 3, cb = (tid & 7) * 32;
        const float4* src =
            (const float4*)(inter + (size_t)ts_ent[row] * I_DIM + kc + cb);
        float4* dst = (float4*)(&Is[row * XS_LD + cb]);
#pragma unroll
        for (int q = 0; q < 8; ++q) dst[q] = src[q];
      }
#endif
      __syncthreads();
      for (int kk = 0; kk < 256; kk += 4) {
        const float* a0p = &Is[lm * XS_LD + kk + lk];
        const float* a1p = &Is[(lm + 16) * XS_LD + kk + lk];
        v2f a0; a0.x = a0p[0]; a0.y = a0p[1];
        v2f a1; a1.x = a1p[0]; a1.y = a1p[1];
        const int kg = kc + kk;
#pragma unroll
        for (int nt = 0; nt < 4; ++nt) {
          v2f b = load_b_frag(Wd, H_DIM, kg, cbase + nt * 16);
          acc[0][nt] = wmma4(a0, b, acc[0][nt]);
          acc[1][nt] = wmma4(a1, b, acc[1][nt]);
        }
      }
    }
#pragma unroll
    for (int mt = 0; mt < 2; ++mt) {
#pragma unroll
      for (int v = 0; v < 8; ++v) {
        const int m = mt * 16 + v + (hi << 3);
        if (ts_val[m]) {
          const float wt = ts_w[m];
          float* op = ypart + (size_t)ts_ent[m] * H_DIM;
#pragma unroll
          for (int nt = 0; nt < 4; ++nt)
            op[cbase + nt * 16 + lm] = acc[mt][nt][v] * wt;
        }
      }
    }
  }
}

// ---------------------------------------------------------------- combine
__global__ void combine_kernel(const float* __restrict__ yp,
                               float* __restrict__ out) {
  const int i = blockIdx.x * blockDim.x + threadIdx.x;   // over T*H/4
  const int hv = H_DIM / 4;
  const int t = i / hv, r = i % hv;
  const float4* a = (const float4*)yp;
  float4* o = (float4*)out;
  const float4 p = a[(size_t)(t * 2 + 0) * hv + r];
  const float4 q = a[(size_t)(t * 2 + 1) * hv + r];
  float4 s;
  s.x = p.x + q.x; s.y = p.y + q.y; s.z = p.z + q.z; s.w = p.w + q.w;
  o[i] = s;
}

// ---------------------------------------------------------------- launch
extern "C" void kernel_launch(void* const* d_in, const int* in_sizes, int n_in,
                              void* d_out, int out_size, void* d_ws, size_t ws_size,
                              hipStream_t stream) {
  const float* x   = (const float*)d_in[0];   // [1,1024,2048]
  const float* gw  = (const float*)d_in[1];   // [16,2048]
  const float* wgu = (const float*)d_in[2];   // [16,2048,2048]
  const float* wd  = (const float*)d_in[3];   // [16,1024,2048]
  float* out = (float*)d_out;                 // [1,1024,2048]

  char* ws = (char*)d_ws;
  int*   cnt     = (int*)ws;                                   // 64 B (pad 256)
  int*   toklist = (int*)(ws + 256);                           // 64 KB
  float* twlist  = (float*)(ws + 256 + 65536);                 // 64 KB
  float* inter   = (float*)(ws + 131584);                      // 2048*1024 f32 = 8 MB
  float* ypart   = (float*)(ws + 131584 + 8388608);            // 2048*2048 f32 = 16 MB

  init_kernel<<<1, 32, 0, stream>>>(cnt);
  router_kernel<<<T_TOK, 256, 0, stream>>>(x, gw, cnt, toklist, twlist);

  dim3 eg(T_TOK / M_TILE, E_NUM);             // worst case: one expert gets all tokens
  moe_gu_kernel<<<eg, 256, 0, stream>>>(x, wgu, cnt, toklist, inter);
  moe_down_kernel<<<eg, 256, 0, stream>>>(inter, wd, cnt, toklist, twlist, ypart);

  combine_kernel<<<(T_TOK * H_DIM / 4) / 256, 256, 0, stream>>>(ypart, out);
}